// BOB_87600152969626
// MI455X (gfx1250) — compile-verified
//
#include <hip/hip_runtime.h>
#include <hip/hip_bf16.h>

typedef __attribute__((ext_vector_type(2))) float v2f;
typedef __attribute__((ext_vector_type(8))) float v8f;

#define NPIX   25088          // 32*28*28
#define NCODE  2048
#define CDIM   384
#define HW     784
#define NB     32
#define FELEM  9633792        // NPIX*CDIM

// ---------------- K1: NCHW -> NHWC transpose (writes output f) ----------------
__global__ void vq_transpose_kernel(const float* __restrict__ feat, float* __restrict__ f) {
    __shared__ float tile[32][33];
    const int b  = blockIdx.z;
    const int c0 = blockIdx.y * 32;
    const int p0 = blockIdx.x * 32;
    const int tx = threadIdx.x & 31;
    const int ty = threadIdx.x >> 5;   // 0..7
    const float* src = feat + ((size_t)b * CDIM + c0) * HW;
    #pragma unroll
    for (int r = ty; r < 32; r += 8) {
        int p = p0 + tx;
        tile[r][tx] = (p < HW) ? src[(size_t)r * HW + p] : 0.0f;
    }
    __syncthreads();
    float* dst = f + ((size_t)b * HW + p0) * CDIM + c0;
    #pragma unroll
    for (int r = ty; r < 32; r += 8) {
        if (p0 + r < HW) dst[(size_t)r * CDIM + tx] = tile[tx][r];
    }
}

// ---------------- K2: per-row L2 norm of f -> inv_norm, x2 ----------------
__global__ void vq_rownorm_kernel(const float* __restrict__ f,
                                  float* __restrict__ inv_norm,
                                  float* __restrict__ x2) {
    __shared__ float sred[128];
    const int n = blockIdx.x;
    const int t = threadIdx.x;                 // 0..127
    const float* row = f + (size_t)n * CDIM;
    float s = 0.0f;
    #pragma unroll
    for (int i = 0; i < 3; ++i) { float a = row[t + 128 * i]; s += a * a; }
    sred[t] = s; __syncthreads();
    for (int st = 64; st > 0; st >>= 1) {
        if (t < st) sred[t] += sred[t + st];
        __syncthreads();
    }
    if (t == 0) {
        float ss  = sred[0];
        float inv = 1.0f / fmaxf(sqrtf(ss), 1e-12f);
        inv_norm[n] = inv;
        x2[n] = ss * inv * inv;
    }
}

// ---------------- K3: scale + transpose to pair-interleaved A^T: At2[c/2][n][2] ----------------
__global__ void vq_scaleT_kernel(const float* __restrict__ feat,
                                 const float* __restrict__ inv_norm,
                                 float* __restrict__ At2) {
    size_t i = (size_t)blockIdx.x * blockDim.x + threadIdx.x;   // over FELEM/2
    if (i >= (size_t)FELEM / 2) return;
    int hw = (int)(i % HW);
    int cp = (int)((i / HW) % (CDIM / 2));
    int b  = (int)(i / ((size_t)HW * (CDIM / 2)));
    int n  = b * HW + hw;
    float inv = inv_norm[n];
    const float* src = feat + ((size_t)b * CDIM + 2 * cp) * HW + hw;
    v2f o;
    o.x = src[0]  * inv;
    o.y = src[HW] * inv;
    *(v2f*)(At2 + ((size_t)cp * NPIX + n) * 2) = o;
}

// ---------------- K4: codebook normalize -> pair-interleaved B^T: Bt2[c/2][k][2] + e2 ----------------
__global__ void vq_codebook_kernel(const float* __restrict__ codebook,
                                   float* __restrict__ Bt2,
                                   float* __restrict__ e2) {
    const int k    = blockIdx.x * 4 + (threadIdx.x >> 5);  // 4 waves/block, 1 row/wave
    const int lane = threadIdx.x & 31;
    const float* row = codebook + (size_t)k * CDIM;
    float vals[12]; float s = 0.0f;
    #pragma unroll
    for (int j = 0; j < 12; ++j) { vals[j] = row[lane + 32 * j]; s += vals[j] * vals[j]; }
    #pragma unroll
    for (int off = 16; off > 0; off >>= 1) s += __shfl_xor(s, off, 32);
    float inv = 1.0f / fmaxf(sqrtf(s), 1e-12f);
    if (lane == 0) e2[k] = s * inv * inv;
    #pragma unroll
    for (int j = 0; j < 12; ++j) {
        int c = lane + 32 * j;
        Bt2[((size_t)(c >> 1) * NCODE + k) * 2 + (c & 1)] = vals[j] * inv;
    }
}

// ---------------- K5: WMMA FP32 GEMM -> distance = x2 + e2 - 2*(A @ B^T) ----------------
// Block: 256 threads = 8 waves; wave tile 32x64 (2x4 WMMA 16x16); block tile 128x128.
__global__ __launch_bounds__(256)
void vq_wmma_dist_kernel(const float* __restrict__ At2,   // [CDIM/2][NPIX][2]
                         const float* __restrict__ Bt2,   // [CDIM/2][NCODE][2]
                         const float* __restrict__ x2,
                         const float* __restrict__ e2,
                         float* __restrict__ dist) {      // [NPIX][NCODE]
    const int lane = threadIdx.x & 31;
    const int wave = threadIdx.x >> 5;       // 0..7
    const int wm   = wave >> 1;              // 0..3
    const int wn   = wave & 1;               // 0..1
    const int lr   = lane & 15;
    const int half = lane >> 4;              // selects K sub-pair per ISA fp32 A/B layout
    const int rowBase = blockIdx.x * 128 + wm * 32;
    const int colBase = blockIdx.y * 128 + wn * 64;

    const int r0 = rowBase + lr, r1 = r0 + 16;
    int cIdx[4];
    #pragma unroll
    for (int nt = 0; nt < 4; ++nt) cIdx[nt] = colBase + nt * 16 + lr;

    v8f acc[2][4] = {};

    for (int kk = 0; kk < CDIM; kk += 4) {
        const size_t kp = (size_t)(kk >> 1) + half;       // K-pair index for this lane half
        const float* ap = At2 + kp * (2 * NPIX);
        const float* bp = Bt2 + kp * (2 * NCODE);
        v2f a0 = *(const v2f*)(ap + 2 * r0);              // global_load_b64, coalesced
        v2f a1 = *(const v2f*)(ap + 2 * r1);
        v2f b[4];
        #pragma unroll
        for (int nt = 0; nt < 4; ++nt) b[nt] = *(const v2f*)(bp + 2 * cIdx[nt]);
        #pragma unroll
        for (int nt = 0; nt < 4; ++nt)
            acc[0][nt] = __builtin_amdgcn_wmma_f32_16x16x4_f32(false, a0, false, b[nt], (short)0, acc[0][nt], false, false);
        #pragma unroll
        for (int nt = 0; nt < 4; ++nt)
            acc[1][nt] = __builtin_amdgcn_wmma_f32_16x16x4_f32(false, a1, false, b[nt], (short)0, acc[1][nt], false, false);
    }

    // Epilogue: C/D layout — VGPR v holds (M = v + 8*half), N = lr (per 16x16 tile)
    #pragma unroll
    for (int mt = 0; mt < 2; ++mt) {
        float x2r[8];
        #pragma unroll
        for (int v = 0; v < 8; ++v) x2r[v] = x2[rowBase + mt * 16 + half * 8 + v];
        #pragma unroll
        for (int nt = 0; nt < 4; ++nt) {
            const int ccol = cIdx[nt];
            const float e2v = e2[ccol];
            #pragma unroll
            for (int v = 0; v < 8; ++v) {
                const int rrow = rowBase + mt * 16 + half * 8 + v;
                dist[(size_t)rrow * NCODE + ccol] = x2r[v] + e2v - 2.0f * acc[mt][nt][v];
            }
        }
    }
}

// ---------------- K6: per-row argmin + softmax(-d/0.1) -> assignment (b,K,h,w), idx ----------------
__global__ __launch_bounds__(256)
void vq_softmax_row_kernel(const float* __restrict__ dist,
                           float* __restrict__ assign,
                           int* __restrict__ idx_out) {
    __shared__ float smin[256];
    __shared__ int   sidx[256];
    const int n  = blockIdx.x;        // 0..NPIX-1
    const int t  = threadIdx.x;       // 0..255
    const int b  = n / HW;
    const int hw = n % HW;
    const float* row = dist + (size_t)n * NCODE;

    float v[8];
    float lmin = 3.4e38f; int lidx = 0;
    #pragma unroll
    for (int i = 0; i < 8; ++i) {
        int k = t + 256 * i;
        v[i] = row[k];
        if (v[i] < lmin) { lmin = v[i]; lidx = k; }
    }
    smin[t] = lmin; sidx[t] = lidx;
    __syncthreads();
    for (int s = 128; s > 0; s >>= 1) {
        if (t < s) {
            float om = smin[t + s]; int oi = sidx[t + s];
            if (om < smin[t] || (om == smin[t] && oi < sidx[t])) { smin[t] = om; sidx[t] = oi; }
        }
        __syncthreads();
    }
    const float dmin = smin[0];
    if (t == 0) idx_out[n] = sidx[0];
    __syncthreads();

    float e[8]; float lsum = 0.0f;
    #pragma unroll
    for (int i = 0; i < 8; ++i) { e[i] = expf((dmin - v[i]) * 10.0f); lsum += e[i]; }
    smin[t] = lsum; __syncthreads();
    for (int s = 128; s > 0; s >>= 1) {
        if (t < s) smin[t] += smin[t + s];
        __syncthreads();
    }
    const float inv = 1.0f / smin[0];
    float* out = assign + (size_t)b * NCODE * HW + hw;
    #pragma unroll
    for (int i = 0; i < 8; ++i) {
        int k = t + 256 * i;
        out[(size_t)k * HW] = e[i] * inv;
    }
}

// ---------------- K7: q_feat[b][c][hw] = code_n[idx[n]][c] (NCHW) ----------------
__global__ void vq_gather_kernel(const float* __restrict__ Bt2,
                                 const int* __restrict__ idx,
                                 float* __restrict__ qf) {
    size_t i = (size_t)blockIdx.x * blockDim.x + threadIdx.x;
    if (i >= (size_t)FELEM) return;
    int hw = (int)(i % HW);
    int c  = (int)((i / HW) % CDIM);
    int b  = (int)(i / ((size_t)HW * CDIM));
    int n  = b * HW + hw;
    qf[i] = Bt2[((size_t)(c >> 1) * NCODE + idx[n]) * 2 + (c & 1)];
}

extern "C" void kernel_launch(void* const* d_in, const int* in_sizes, int n_in,
                              void* d_out, int out_size, void* d_ws, size_t ws_size,
                              hipStream_t stream) {
    const float* feat     = (const float*)d_in[0];   // (32,384,28,28)
    const float* codebook = (const float*)d_in[1];   // (2048,384)

    float* out       = (float*)d_out;
    float* out_f     = out;                                   // FELEM
    float* out_qf    = out + (size_t)FELEM;                   // FELEM
    float* out_asg   = out + (size_t)2 * FELEM;               // NPIX*NCODE
    float* out_dist  = out + (size_t)2 * FELEM + (size_t)NPIX * NCODE;

    float* ws   = (float*)d_ws;
    float* At2  = ws;                                 // CDIM*NPIX  = 9,633,792
    float* Bt2  = At2 + (size_t)CDIM * NPIX;          // CDIM*NCODE =   786,432
    float* x2   = Bt2 + (size_t)CDIM * NCODE;         // NPIX
    float* e2   = x2 + NPIX;                          // NCODE
    float* invn = e2 + NCODE;                         // NPIX
    int*   idx  = (int*)(invn + NPIX);                // NPIX ints

    // K1: transpose feat -> f (output 0)
    vq_transpose_kernel<<<dim3((HW + 31) / 32, CDIM / 32, NB), 256, 0, stream>>>(feat, out_f);
    // K2: row norms of f
    vq_rownorm_kernel<<<NPIX, 128, 0, stream>>>(out_f, invn, x2);
    // K3: normalized, pair-interleaved K-major A^T
    vq_scaleT_kernel<<<(FELEM / 2 + 255) / 256, 256, 0, stream>>>(feat, invn, At2);
    // K4: normalized, pair-interleaved K-major B^T + e2
    vq_codebook_kernel<<<NCODE / 4, 128, 0, stream>>>(codebook, Bt2, e2);
    // K5: WMMA fp32 distance GEMM (output 3)
    vq_wmma_dist_kernel<<<dim3(NPIX / 128, NCODE / 128), 256, 0, stream>>>(At2, Bt2, x2, e2, out_dist);
    // K6: argmin + softmax -> assignment (output 2), idx
    vq_softmax_row_kernel<<<NPIX, 256, 0, stream>>>(out_dist, out_asg, idx);
    // K7: gather nearest code -> q_feat (output 1)
    vq_gather_kernel<<<(FELEM + 255) / 256, 256, 0, stream>>>(Bt2, idx, out_qf);
}